// GraphSAGEConv_47339129536946
// MI455X (gfx1250) — compile-verified
//
#include <hip/hip_runtime.h>
#include <hip/hip_bf16.h>

// GraphSAGE conv for MI455X (gfx1250, wave32).
//   agg[dst] += x[src] over E edges (L2-resident f32 atomics; agg = 51MB < 192MB L2)
//   out = relu([x | agg/max(deg,1)] @ W^T + b) via V_WMMA_F32_16X16X4_F32
//
// d_ws layout: agg = float[N*128] at offset 0, deg = float[N] after it (~51.6 MB).

typedef __attribute__((ext_vector_type(2))) float v2f;
typedef __attribute__((ext_vector_type(8))) float v8f;

#define D_IN   128
#define KDIM   256     // 2*D_IN (concat)
#define OUTF   128
#define TILE_M 16
#define LDS_PITCH 260  // 256 + 4 pad: rows land 4 banks apart -> conflict-free frag reads

// ---------------------------------------------------------------- zero init
__global__ void sage_zero_kernel(float4* __restrict__ p, long long n4) {
    long long i = (long long)blockIdx.x * blockDim.x + threadIdx.x;
    if (i < n4) p[i] = make_float4(0.f, 0.f, 0.f, 0.f);
}

// ---------------------------------------------------------------- edge scatter
// One wave32 per edge: lane l covers features [4l, 4l+4).
__global__ __launch_bounds__(256)
void sage_scatter_kernel(const float* __restrict__ x,
                         const long long* __restrict__ edge_index,
                         float* __restrict__ agg, float* __restrict__ deg,
                         int nEdges) {
    int gt   = blockIdx.x * blockDim.x + threadIdx.x;
    int edge = gt >> 5;
    int lane = gt & 31;
    if (edge >= nEdges) return;

    long long src = edge_index[edge];                      // edge_index[0][e]
    long long dst = edge_index[(long long)nEdges + edge];  // edge_index[1][e]

    const float4* xs = (const float4*)(x + src * (long long)D_IN);
    float4 v = xs[lane];                                   // global_load_b128

    float* ad = agg + dst * (long long)D_IN + lane * 4;
    atomicAdd(ad + 0, v.x);                                // global_atomic_add_f32 (L2)
    atomicAdd(ad + 1, v.y);
    atomicAdd(ad + 2, v.z);
    atomicAdd(ad + 3, v.w);
    if (lane == 0) atomicAdd(&deg[dst], 1.0f);
}

// ---------------------------------------------------------------- fused GEMM
// Block: 256 threads = 8 waves. Block owns a 16-node M-tile; wave w owns
// output columns [16w, 16w+16). K = 256 swept in 64 steps of V_WMMA_F32_16X16X4_F32.
//
// ISA f32 fragment layouts (CDNA5 ISA §7.12.2):
//   A (16x4):  v0 = {lanes0-15: K=0, lanes16-31: K=2}, v1 = {K=1, K=3}, M = lane&15
//   B (4x16):  v0 = {lanes0-15: row K=0, lanes16-31: row K=2}, v1 = {K=1, K=3}, N = lane&15
//   C/D (16x16): VGPR v = rows {v, v+8} across the two lane halves, N = lane&15
__global__ __launch_bounds__(256)
void sage_gemm_kernel(const float* __restrict__ x,
                      const float* __restrict__ W,      // [OUTF][KDIM]
                      const float* __restrict__ bias,   // [OUTF]
                      const float* __restrict__ agg,
                      const float* __restrict__ deg,
                      float* __restrict__ out) {
    __shared__ float cat[TILE_M][LDS_PITCH];   // ~16.6 KB of the 320 KB WGP LDS

    const int mbase = blockIdx.x * TILE_M;
    const int t     = threadIdx.x;

    // Stage cat[16][256] = [x | agg*invdeg]; thread t -> row t>>4, 16-col slab.
    {
        int r    = t >> 4;
        int c0   = (t & 15) * 16;
        long long node = mbase + r;
        if (c0 < D_IN) {
            const float4* p = (const float4*)(x + node * D_IN + c0);
#pragma unroll
            for (int j = 0; j < 4; ++j) {
                float4 v = p[j];
                float* dptr = &cat[r][c0 + j * 4];
                dptr[0] = v.x; dptr[1] = v.y; dptr[2] = v.z; dptr[3] = v.w;
            }
        } else {
            float inv = 1.0f / fmaxf(deg[node], 1.0f);
            const float4* p = (const float4*)(agg + node * D_IN + (c0 - D_IN));
#pragma unroll
            for (int j = 0; j < 4; ++j) {
                float4 v = p[j];
                float* dptr = &cat[r][c0 + j * 4];
                dptr[0] = v.x * inv; dptr[1] = v.y * inv;
                dptr[2] = v.z * inv; dptr[3] = v.w * inv;
            }
        }
    }
    __syncthreads();

    const int wave = t >> 5;
    const int lane = t & 31;
    const int hi   = lane >> 4;          // 0: K pair {0,1}; 1: K pair {2,3}
    const int lo   = lane & 15;          // A row (M) / B+D column (N)
    const int ncol = wave * 16 + lo;     // output column this lane produces
    const float* wrow = W + (long long)ncol * KDIM;  // B[k][ncol] = W[ncol][k]

    v8f acc = {};
#pragma unroll 8
    for (int k0 = 0; k0 < KDIM; k0 += 4) {
        int kb = k0 + hi * 2;
        v2f a, bm;
        a.x  = cat[lo][kb];              // ds reads (b64-pairable, bank-spread)
        a.y  = cat[lo][kb + 1];
        bm.x = wrow[kb];                 // global b64, W is L2/WGP$-hot
        bm.y = wrow[kb + 1];
        acc = __builtin_amdgcn_wmma_f32_16x16x4_f32(
            /*neg_a=*/false, a, /*neg_b=*/false, bm,
            /*c_mod=*/(short)0, acc, /*reuse_a=*/false, /*reuse_b=*/false);
    }

    const float bv = bias[ncol];
#pragma unroll
    for (int v = 0; v < 8; ++v) {
        int row = v + hi * 8;            // D-matrix row of VGPR v for this lane half
        long long node = mbase + row;
        out[node * (long long)OUTF + ncol] = fmaxf(acc[v] + bv, 0.0f);
    }
}

// ---------------------------------------------------------------- launcher
extern "C" void kernel_launch(void* const* d_in, const int* in_sizes, int n_in,
                              void* d_out, int out_size, void* d_ws, size_t ws_size,
                              hipStream_t stream) {
    const float*     x    = (const float*)d_in[0];
    const long long* ei   = (const long long*)d_in[1];   // int64 [2][E]
    const float*     W    = (const float*)d_in[2];
    const float*     bias = (const float*)d_in[3];
    float*           out  = (float*)d_out;

    const int nNodes = in_sizes[0] / D_IN;   // 100000
    const int nEdges = in_sizes[1] / 2;      // 600000

    float* agg = (float*)d_ws;
    float* deg = agg + (long long)nNodes * D_IN;

    // 1) zero agg + deg (N*129 floats, divisible by 4 -> pure float4 stores)
    long long n4 = ((long long)nNodes * (D_IN + 1)) / 4;
    int zb = (int)((n4 + 255) / 256);
    sage_zero_kernel<<<zb, 256, 0, stream>>>((float4*)d_ws, n4);

    // 2) scatter-add: one wave32 per edge
    long long sthreads = (long long)nEdges * 32;
    int sb = (int)((sthreads + 255) / 256);
    sage_scatter_kernel<<<sb, 256, 0, stream>>>(x, ei, agg, deg, nEdges);

    // 3) fused concat + GEMM + bias + relu  (100000 % 16 == 0 -> no tail)
    sage_gemm_kernel<<<nNodes / TILE_M, 256, 0, stream>>>(x, W, bias, agg, deg, out);
}